// GCNlayer_56693568307362
// MI455X (gfx1250) — compile-verified
//
#include <hip/hip_runtime.h>

typedef __attribute__((ext_vector_type(2))) float v2f;
typedef __attribute__((ext_vector_type(4))) float v4f;
typedef __attribute__((ext_vector_type(8))) float v8f;

#define DFEAT 64

// ---------------------------------------------------------------------------
// Kernel 1: zero the temp accumulator (must happen every call; harness replays)
// ---------------------------------------------------------------------------
__global__ void gcn_zero_kernel(v4f* __restrict__ p, int n4) {
  int i = blockIdx.x * blockDim.x + threadIdx.x;
  if (i < n4) {
    v4f z = {0.0f, 0.0f, 0.0f, 0.0f};
    p[i] = z;
  }
}

// ---------------------------------------------------------------------------
// Kernel 2: COO SpMM scatter.  16 threads per edge, float4 feature chunk each.
// X (25.6MB) is L2-resident (192MB L2), so gathers are L2 hits; the 102.4M
// f32 scatter-adds land as L2-side atomics on the L2-resident temp buffer.
// ---------------------------------------------------------------------------
__global__ void gcn_scatter_kernel(const float* __restrict__ X,
                                   const float* __restrict__ vals,
                                   const int* __restrict__ row,
                                   const int* __restrict__ col,
                                   float* __restrict__ temp,
                                   int n_edges) {
  long long t = (long long)blockIdx.x * blockDim.x + threadIdx.x;
  int e = (int)(t >> 4);
  if (e >= n_edges) return;
  int chunk = ((int)t & 15) * 4;

  int r = row[e];
  int c = col[e];
  float v = vals[e];

  v4f x = *(const v4f*)(X + (long long)c * DFEAT + chunk);
  float* td = temp + (long long)r * DFEAT + chunk;
  atomicAdd(td + 0, x.x * v);
  atomicAdd(td + 1, x.y * v);
  atomicAdd(td + 2, x.z * v);
  atomicAdd(td + 3, x.w * v);
}

// ---------------------------------------------------------------------------
// Kernel 3: out = temp @ W + b using V_WMMA_F32_16X16X4_F32.
// One wave32 computes a 16(M) x 64(N) strip: 16 K-steps x 4 N-tiles = 64 WMMAs.
//
// W is staged in LDS *K-pair interleaved*:  Wl2[(k>>1)*128 + n*2 + (k&1)]
// so each lane's B operand (W[k][n], W[k+1][n]) is a contiguous 8B-aligned
// float2 -> single ds_load_b64, no register shuffles before the WMMA.
//
// ISA VGPR layouts used (CDNA5 ISA, 32-bit operands):
//   A 16x4 : lane&15 = M row; lanes 0-15 hold K={0,1}, lanes 16-31 K={2,3}
//   B 4x16 : lane&15 = N col; lanes 0-15 hold K={0,1}, lanes 16-31 K={2,3}
//   C/D    : VGPR r -> M = r + 8*(lane>>4), N = lane&15
// ---------------------------------------------------------------------------
__global__ void __launch_bounds__(256)
gcn_gemm_wmma_kernel(const float* temp,            // may alias `out` (fallback)
                     const float* __restrict__ W,
                     const float* __restrict__ bias,
                     float* out, int n_rows) {
  __shared__ float Wl2[DFEAT * DFEAT];  // pair-interleaved layout
  __shared__ float bl[DFEAT];
  for (int i = threadIdx.x; i < DFEAT * DFEAT; i += 256) {
    int k = i >> 6;          // row of W
    int n = i & 63;          // col of W
    Wl2[(k >> 1) * 128 + n * 2 + (k & 1)] = W[i];
  }
  if (threadIdx.x < DFEAT) bl[threadIdx.x] = bias[threadIdx.x];
  __syncthreads();

  int wave = (int)((blockIdx.x * 256u + threadIdx.x) >> 5);  // wave32
  int lane = (int)(threadIdx.x & 31u);
  int m0 = wave * 16;
  if (m0 >= n_rows) return;  // wave-uniform: EXEC stays all-1s for WMMA

  int lo = lane & 15;        // M (for A) / N (for B,C,D)
  int hi = lane >> 4;        // half-wave selector
  int koff = hi * 2;         // K sub-offset within a K=4 step

  // Accumulators for 4 N-tiles, bias folded in (bias depends only on N).
  v8f acc[4];
#pragma unroll
  for (int t = 0; t < 4; ++t) {
    float bv = bl[t * 16 + lo];
#pragma unroll
    for (int r = 0; r < 8; ++r) acc[t][r] = bv;
  }

  // Clamp A-row for ragged tails (stores are guarded below).
  int mrow = m0 + lo;
  if (mrow >= n_rows) mrow = n_rows - 1;
  const float* arow = temp + (long long)mrow * DFEAT;

  // Per-lane LDS base for B: pair index (k>>1), column (t*16+lo).
  // Element float2 index within a K-step block: n*1 pairs of floats.
  const v2f* wl2 = (const v2f*)Wl2;   // 2048 float2 pairs

#pragma unroll
  for (int kk = 0; kk < 16; ++kk) {
    int kb = kk * 4;
    int kpair = (kb + koff) >> 1;     // (k)/2 with k = kb+koff (koff even)
    v2f a;
    a.x = arow[kb + koff];
    a.y = arow[kb + koff + 1];
#pragma unroll
    for (int t = 0; t < 4; ++t) {
      v2f bb = wl2[kpair * 64 + t * 16 + lo];   // single ds_load_b64
      acc[t] = __builtin_amdgcn_wmma_f32_16x16x4_f32(
          /*neg_a=*/false, a, /*neg_b=*/false, bb,
          /*c_mod=*/(short)0, acc[t], /*reuse_a=*/false, /*reuse_b=*/false);
    }
  }

#pragma unroll
  for (int t = 0; t < 4; ++t) {
#pragma unroll
    for (int r = 0; r < 8; ++r) {
      int m = m0 + r + 8 * hi;
      if (m < n_rows) out[(long long)m * DFEAT + t * 16 + lo] = acc[t][r];
    }
  }
}

// ---------------------------------------------------------------------------
extern "C" void kernel_launch(void* const* d_in, const int* in_sizes, int n_in,
                              void* d_out, int out_size, void* d_ws, size_t ws_size,
                              hipStream_t stream) {
  // setup_inputs order: X, vals, W, b, row, col
  const float* X    = (const float*)d_in[0];
  const float* vals = (const float*)d_in[1];
  const float* W    = (const float*)d_in[2];
  const float* bias = (const float*)d_in[3];
  const int*   row  = (const int*)d_in[4];
  const int*   col  = (const int*)d_in[5];
  float* out = (float*)d_out;

  int n_nodes = in_sizes[0] / DFEAT;
  int n_edges = in_sizes[1];

  size_t temp_bytes = (size_t)n_nodes * DFEAT * sizeof(float);
  // Prefer workspace; fall back to in-place accumulation in d_out (safe: each
  // GEMM wave reads only its own 16 rows and stores strictly after its loads).
  float* temp = (ws_size >= temp_bytes) ? (float*)d_ws : out;

  // 1) zero temp
  int n4 = n_nodes * (DFEAT / 4);
  gcn_zero_kernel<<<(n4 + 255) / 256, 256, 0, stream>>>((v4f*)temp, n4);

  // 2) edge scatter (16 threads per edge)
  long long sthreads = (long long)n_edges * 16;
  unsigned sblocks = (unsigned)((sthreads + 255) / 256);
  gcn_scatter_kernel<<<sblocks, 256, 0, stream>>>(X, vals, row, col, temp, n_edges);

  // 3) WMMA GEMM + bias
  int waves = (n_nodes + 15) / 16;
  long long gthreads = (long long)waves * 32;
  unsigned gblocks = (unsigned)((gthreads + 255) / 256);
  gcn_gemm_wmma_kernel<<<gblocks, 256, 0, stream>>>(temp, W, bias, out, n_nodes);
}